// RelationLoss_57913339019396
// MI455X (gfx1250) — compile-verified
//
#include <hip/hip_runtime.h>
#include <hip/hip_bf16.h>
#include <math.h>

// Problem constants from the reference
#define RB   8
#define RR   50
#define RC   150
#define RH   128
#define RW   128
#define RN   4096
#define RK   192
#define LOSS_WEIGHT 0.1f

// ---------------------------------------------------------------------------
// Kernel 1: one wave32 per relation. Lane L owns samples L, L+32, ... (6 each).
// Duplicate test recomputes sample k-1 analytically, so no cross-lane comm in
// the gather loop. Wave reduction of (sum_dots, count) via shfl_xor (wave32).
// Writes logp for each relation into ws[r].
// ---------------------------------------------------------------------------
__global__ void relation_loss_per_rel(const float* __restrict__ rafs,
                                      const float* __restrict__ heat,
                                      const int*   __restrict__ rels,
                                      float*       __restrict__ ws) {
    const int wave = threadIdx.x >> 5;            // 8 waves per 256-thread block
    const int lane = threadIdx.x & 31;
    const int r    = blockIdx.x * 8 + wave;       // 512 blocks * 8 = 4096
    if (r >= RN) return;

    const int* rel = rels + (size_t)r * 8;
    const int bi   = rel[0];
    const int scls = rel[1];
    const int sy   = rel[2];
    const int sx   = rel[3];
    const int ocls = rel[4];
    const int oy   = rel[5];
    const int ox   = rel[6];
    const int pred = rel[7];

    const float subj = heat[(((size_t)bi * RC + scls) * RH + sy) * RW + sx];
    const float obj  = heat[(((size_t)bi * RC + ocls) * RH + oy) * RW + ox];

    const float dx   = (float)(ox - sx);
    const float dy   = (float)(oy - sy);
    const float norm = sqrtf(dx * dx + dy * dy);
    const float ux   = dx / norm;
    const float uy   = dy / norm;
    const int   num  = (int)ceilf(norm);          // >= 1 (distinct endpoints)
    const float inv  = 1.0f / (float)((num - 1) > 1 ? (num - 1) : 1);

    const float fox = (float)ox, foy = (float)oy;
    const float ddx = (float)(sx - ox);
    const float ddy = (float)(sy - oy);

    // raf channels: rafs viewed as (B, R, 2, H, W) -> plane (bi, pred, ch)
    const float* rafb = rafs + (((size_t)bi * (2 * RR)) + (size_t)(2 * pred)) * (RH * RW);

    float sum = 0.0f;
    float cnt = 0.0f;
    #pragma unroll
    for (int k = lane; k < RK; k += 32) {
        if (k >= num) continue;                   // in_len mask
        const float t  = (float)k * inv;
        const int   px = (int)rintf(fox + t * ddx);
        const int   py = (int)rintf(foy + t * ddy);
        bool dup = false;
        if (k > 0) {
            const float tp  = (float)(k - 1) * inv;
            const int   pxp = (int)rintf(fox + tp * ddx);
            const int   pyp = (int)rintf(foy + tp * ddy);
            dup = (px == pxp) && (py == pyp);
        }
        if (!dup) {
            const int off = py * RW + px;         // in-bounds for k < num
            float vx = rafb[off];
            float vy = rafb[RH * RW + off];
            vx = fminf(fmaxf(vx, -1.0f), 1.0f);   // clip(rafs, -1, 1)
            vy = fminf(fmaxf(vy, -1.0f), 1.0f);
            sum += vx * ux + vy * uy;
            cnt += 1.0f;
        }
    }

    // wave32 reduction (fixed order -> deterministic)
    #pragma unroll
    for (int m = 16; m >= 1; m >>= 1) {
        sum += __shfl_xor(sum, m, 32);
        cnt += __shfl_xor(cnt, m, 32);
    }

    float integral = sum / cnt;                   // cnt >= 1 (k=0 always valid)
    integral = fminf(fmaxf(integral, 0.0f), 1.0f);
    const float score = subj * obj * integral;
    const float logp  = logf(fmaxf(score, 1e-12f));

    if (lane == 0) ws[r] = logp;
}

// ---------------------------------------------------------------------------
// Kernel 2: single wave32 reduces 4096 floats with V_WMMA_F32_16X16X4_F32.
// B = all-ones  =>  D[m][n] = sum_k A[m][k] + C[m][n]: each WMMA folds 64 new
// values (A operand) into the C accumulator. Because B is all ones, the A/B
// lane layout is irrelevant to the total. Grand total = sum over M of any one
// column of D; column N=0 lives in lane 0 (M=0..7 in v0..v7) and lane 16
// (M=8..15 in v0..v7).
// ---------------------------------------------------------------------------
typedef float v2f __attribute__((ext_vector_type(2)));
typedef float v8f __attribute__((ext_vector_type(8)));

__global__ void reduce_loss_wmma(const float* __restrict__ ws,
                                 float*       __restrict__ out) {
    const int lane = threadIdx.x;                 // 0..31, EXEC all ones
    v8f c = {};                                   // C accumulator = 0
    v2f b;  b.x = 1.0f; b.y = 1.0f;               // all-ones B (4x16)

    for (int i = 0; i < RN / 64; ++i) {           // 64 iterations
        const float2 v = *(const float2*)(ws + i * 64 + lane * 2);
        v2f a;  a.x = v.x;  a.y = v.y;            // 64 fresh values per WMMA
        // 8 args: (neg_a, A, neg_b, B, c_mod, C, reuse_a, reuse_b)
        c = __builtin_amdgcn_wmma_f32_16x16x4_f32(
                false, a, false, b, (short)0, c, false, false);
    }

    float total = 0.0f;
    #pragma unroll
    for (int v = 0; v < 8; ++v) {
        total += __shfl(c[v], 0, 32);             // M = v,   N = 0
        total += __shfl(c[v], 16, 32);            // M = v+8, N = 0
    }

    if (lane == 0)
        out[0] = -total * (LOSS_WEIGHT / (float)RN);
}

// ---------------------------------------------------------------------------
extern "C" void kernel_launch(void* const* d_in, const int* in_sizes, int n_in,
                              void* d_out, int out_size, void* d_ws, size_t ws_size,
                              hipStream_t stream) {
    const float* rafs = (const float*)d_in[0];   // (8, 100, 128, 128) f32
    const float* heat = (const float*)d_in[1];   // (8, 150, 128, 128) f32
    const int*   rels = (const int*)d_in[2];     // (4096, 8) i32
    float* ws  = (float*)d_ws;                   // 4096 floats scratch
    float* out = (float*)d_out;                  // scalar f32

    relation_loss_per_rel<<<RN / 8, 256, 0, stream>>>(rafs, heat, rels, ws);
    reduce_loss_wmma<<<1, 32, 0, stream>>>(ws, out);
}